// LocalConvModule_74775380623610
// MI455X (gfx1250) — compile-verified
//
#include <hip/hip_runtime.h>

#define N_B   128
#define C_CH  256
#define HW    1024
#define TOPK  64
// d_out layout (float element offsets), outputs concatenated in return order
#define CONCAT_ROW (TOPK * C_CH + HW)          // 17408
#define ST_OFF     (N_B * CONCAT_ROW)          // 2228224
#define OUT_OFF    (ST_OFF + N_B * HW)         // 2359296

typedef float v2f __attribute__((ext_vector_type(2)));
typedef float v4f __attribute__((ext_vector_type(4)));
typedef float v8f __attribute__((ext_vector_type(8)));

// ---------------------------------------------------------------------------
// Kernel 1: 1x1 conv (contraction over 256 channels) via V_WMMA_F32_16X16X4_F32.
// One wave computes logits for 16 spatial positions; 8 waves/block -> 128
// positions; grid (8, 128). A = x tile (16 pos x 4 ch), B = w chunk replicated
// across all N columns, so every D column holds the same dot product.
// Runs AFTER the zero-fill so x is the freshest data in L2 for the scatter.
// ---------------------------------------------------------------------------
__global__ __launch_bounds__(256) void k_conv_logits(const float* __restrict__ x,
                                                     const float* __restrict__ w,
                                                     float* __restrict__ logits) {
  __shared__ float wLds[C_CH];
  int tid = threadIdx.x;
  wLds[tid] = w[tid];
  __syncthreads();

  int n        = blockIdx.y;
  int wave     = tid >> 5;
  int lane     = tid & 31;
  int laneHalf = lane >> 4;     // 0: K=0,1  1: K=2,3  (16x4 f32 A layout)
  int l        = lane & 15;     // M = row = spatial position within tile
  int pos0     = blockIdx.x * 128 + wave * 16;

  const float* xb = x + (size_t)n * (C_CH * HW) + pos0 + l;

  v8f acc = {};
  for (int c0 = 0; c0 < C_CH; c0 += 4) {
    int ca = c0 + laneHalf * 2;
    v2f a, b;
    a.x = xb[(size_t)ca * HW];          // A[l][K]   K = laneHalf*2 + 0
    a.y = xb[(size_t)(ca + 1) * HW];    // A[l][K+1]
    b.x = wLds[ca];                     // B[K][*]  replicated across N columns
    b.y = wLds[ca + 1];                 // B[K+1][*]
    acc = __builtin_amdgcn_wmma_f32_16x16x4_f32(
        /*neg_a=*/false, a, /*neg_b=*/false, b,
        /*c_mod=*/(short)0, acc, /*reuse_a=*/false, /*reuse_b=*/false);
  }

  // D layout: lane half h, VGPR v -> row M = h*8 + v; all columns equal.
  if (l == 0) {
    float* dst = logits + n * HW + pos0 + laneHalf * 8;
#pragma unroll
    for (int v = 0; v < 8; ++v) dst[v] = acc[v];
  }
}

// ---------------------------------------------------------------------------
// Kernel 2: per-image softmax over 1024 positions, exact jax top-64
// (descending p, ties -> lower index), indices sorted ascending, st written.
// One block of 1024 threads per image.
// ---------------------------------------------------------------------------
__global__ __launch_bounds__(1024) void k_softmax_topk(const float* __restrict__ logits,
                                                       float* __restrict__ dout,
                                                       int* __restrict__ wsIdx) {
  __shared__ float red[HW];
  __shared__ float key[HW];
  __shared__ int   kid[HW];
  __shared__ int   flag[HW];
  __shared__ int   tIdx[TOPK];

  int n   = blockIdx.x;
  int tid = threadIdx.x;

  float lg = logits[n * HW + tid];

  // max reduction
  red[tid] = lg; __syncthreads();
  for (int s = 512; s > 0; s >>= 1) {
    if (tid < s) red[tid] = fmaxf(red[tid], red[tid + s]);
    __syncthreads();
  }
  float mx = red[0]; __syncthreads();

  float e = expf(lg - mx);
  red[tid] = e; __syncthreads();
  for (int s = 512; s > 0; s >>= 1) {
    if (tid < s) red[tid] += red[tid + s];
    __syncthreads();
  }
  float p = e / red[0];
  __syncthreads();

  key[tid]  = p;
  kid[tid]  = tid;
  flag[tid] = 0;
  __syncthreads();

  // bitonic sort: "before" = higher p first, tie -> smaller index first
  for (int k = 2; k <= HW; k <<= 1) {
    for (int j = k >> 1; j > 0; j >>= 1) {
      int ixj = tid ^ j;
      if (ixj > tid) {
        float pa = key[tid], pb = key[ixj];
        int   ia = kid[tid], ib = kid[ixj];
        bool before = (pa > pb) || (pa == pb && ia < ib);
        bool up     = ((tid & k) == 0);
        if (up != before) {
          key[tid] = pb; key[ixj] = pa;
          kid[tid] = ib; kid[ixj] = ia;
        }
      }
      __syncthreads();
    }
  }

  if (tid < TOPK) tIdx[tid] = kid[tid];
  __syncthreads();

  // sort the 64 winning indices ascending
  for (int k = 2; k <= TOPK; k <<= 1) {
    for (int j = k >> 1; j > 0; j >>= 1) {
      if (tid < TOPK) {
        int ixj = tid ^ j;
        if (ixj > tid) {
          int a = tIdx[tid], b = tIdx[ixj];
          bool up = ((tid & k) == 0);
          if (up ? (a > b) : (a < b)) { tIdx[tid] = b; tIdx[ixj] = a; }
        }
      }
      __syncthreads();
    }
  }

  if (tid < TOPK) {
    flag[tIdx[tid]] = 1;
    wsIdx[n * TOPK + tid] = tIdx[tid];
  }
  __syncthreads();

  // st = (mask - p) + p  (exactly 0 at non-selected positions)
  float st = ((float)flag[tid] - p) + p;
  dout[n * CONCAT_ROW + TOPK * C_CH + tid] = st;  // concat_out st part
  dout[ST_OFF + n * HW + tid]              = st;  // st_mask output
}

// ---------------------------------------------------------------------------
// Kernel 3: zero-fill the `out` region (st is exactly 0 off the top-k set,
// so out = 0 everywhere except the 64 selected columns). Pure-store pass with
// NON-TEMPORAL hints so the 134 MB of zeros do not evict x from the 192 MB L2
// (the later scatter kernel's uncoalesced x gathers then hit L2).
// ---------------------------------------------------------------------------
__global__ __launch_bounds__(256) void k_zero_out(v4f* __restrict__ o) {
  size_t i = (size_t)blockIdx.x * 256 + threadIdx.x;
  v4f z = {0.f, 0.f, 0.f, 0.f};
  __builtin_nontemporal_store(z, &o[i]);
}

// ---------------------------------------------------------------------------
// Kernel 4: for each (image, k) write the selected column: gathered
// out_efficient[n,k,c] and the non-zero entries of out[n,c,idx].
// x reads are scattered (4 KB channel stride) but hit L2: x was streamed in by
// k_conv_logits just before, and the zero-fill used NT stores.
// ---------------------------------------------------------------------------
__global__ __launch_bounds__(256) void k_scatter(const float* __restrict__ x,
                                                 const int* __restrict__ wsIdx,
                                                 float* __restrict__ dout) {
  int k = blockIdx.x;   // 0..63
  int n = blockIdx.y;   // 0..127
  int c = threadIdx.x;  // 0..255

  int   sidx = wsIdx[n * TOPK + k];
  float stv  = dout[ST_OFF + n * HW + sidx];
  float v    = stv * x[(size_t)n * (C_CH * HW) + (size_t)c * HW + sidx];

  dout[n * CONCAT_ROW + k * C_CH + c] = v;                                     // out_efficient
  dout[(size_t)OUT_OFF + (size_t)n * (C_CH * HW) + (size_t)c * HW + sidx] = v; // out
}

extern "C" void kernel_launch(void* const* d_in, const int* in_sizes, int n_in,
                              void* d_out, int out_size, void* d_ws, size_t ws_size,
                              hipStream_t stream) {
  (void)in_sizes; (void)n_in; (void)out_size; (void)ws_size;
  const float* x = (const float*)d_in[0];
  const float* w = (const float*)d_in[1];
  float* out = (float*)d_out;

  float* wsLogits = (float*)d_ws;                                     // 128*1024 f32
  int*   wsIdx    = (int*)((char*)d_ws + N_B * HW * sizeof(float));   // 128*64 i32

  // Order chosen for L2 residency: zero-fill (NT, no deps) first, then the
  // conv streams x into L2, then the tiny topk, then the L2-hitting scatter.
  k_zero_out<<<dim3((N_B * C_CH * HW / 4) / 256), 256, 0, stream>>>(
      (v4f*)(out + OUT_OFF));
  k_conv_logits<<<dim3(8, N_B), 256, 0, stream>>>(x, w, wsLogits);
  k_softmax_topk<<<dim3(N_B), 1024, 0, stream>>>(wsLogits, out, wsIdx);
  k_scatter<<<dim3(TOPK, N_B), 256, 0, stream>>>(x, wsIdx, out);
}